// VGAE_80522046866107
// MI455X (gfx1250) — compile-verified
//
#include <hip/hip_runtime.h>
#include <hip/hip_bf16.h>
#include <math.h>

// ---------------------------------------------------------------------------
// VGAE: 3x GCNConv encoder + reparameterize + dense sigmoid(z z^T) decoder.
// N = 8192 nodes, E = 262144 edges, feature widths 256 -> 64 -> (32|32).
// Decoder is the hot loop: 256 MB f32 output, ~4.3 GFLOP. Store floor at
// 23.3 TB/s is ~11.5 us; f32 WMMA time is the same order, so we exploit
// symmetry of z z^T: compute only upper-triangular 128x128 tiles and mirror
// each off-diagonal tile via a per-wave LDS transpose (coalesced both ways).
// All matrix math runs through V_WMMA_F32_16X16X4_F32 (full f32 precision;
// the kernel is bandwidth-bound so lower precision buys nothing).
// ---------------------------------------------------------------------------

typedef float v2f __attribute__((ext_vector_type(2)));
typedef float v8f __attribute__((ext_vector_type(8)));

#define NNODES 8192
#define FMID   64        // width of t1/h/t2/acc buffers

// ----------------------------- tiny utility kernels ------------------------

__global__ void k_fill(float* __restrict__ p, float v, int n) {
    int i = blockIdx.x * blockDim.x + threadIdx.x;
    if (i < n) p[i] = v;
}

__global__ void k_degree(const long long* __restrict__ dst,
                         float* __restrict__ deg, int E) {
    int e = blockIdx.x * blockDim.x + threadIdx.x;
    if (e < E) atomicAdd(&deg[(int)dst[e]], 1.0f);
}

__global__ void k_dinv(const float* __restrict__ deg,
                       float* __restrict__ dinv, int n) {
    int i = blockIdx.x * blockDim.x + threadIdx.x;
    if (i < n) {
        float d = deg[i];
        dinv[i] = d > 0.0f ? rsqrtf(d) : 0.0f;
    }
}

// Pack Wmu [64x32] and Wls [64x32] into Wcat [64x64] so both heads share one
// GEMM and one edge-propagation pass.
__global__ void k_pack_wcat(const float* __restrict__ Wmu,
                            const float* __restrict__ Wls,
                            float* __restrict__ Wcat) {
    int idx = blockIdx.x * blockDim.x + threadIdx.x;   // 64*64
    if (idx < 64 * 64) {
        int k = idx >> 6, j = idx & 63;
        Wcat[idx] = (j < 32) ? Wmu[k * 32 + j] : Wls[k * 32 + (j - 32)];
    }
}

// ----------------------------- WMMA dense GEMM -----------------------------
// C[M,N] = A[M,K] @ B[K,N], row-major, M%64==0, N%16==0, K%4==0.
// Block = 4 waves; wave w computes the 16x16 tile (bx*64 + w*16, by*16).
// A-frag layout (ISA 7.12.2, 32-bit 16x4): lanes 0-15 -> K={k,k+1},
// lanes 16-31 -> K={k+2,k+3}. B-frag mirrors with N across lanes.
__global__ __launch_bounds__(128)
void k_gemm_wmma(const float* __restrict__ A, const float* __restrict__ B,
                 float* __restrict__ C, int M, int N, int K) {
    const int wave = threadIdx.x >> 5;
    const int lane = threadIdx.x & 31;
    const int half = lane >> 4;
    const int lm   = lane & 15;
    const int m0 = blockIdx.x * 64 + wave * 16;
    const int n0 = blockIdx.y * 16;
    if (m0 >= M) return;

    v8f c = {};
    const float* arow = A + (size_t)(m0 + lm) * K + 2 * half;
    const float* bcol = B + (size_t)(2 * half) * N + (n0 + lm);
    for (int k = 0; k < K; k += 4) {
        v2f a, b;
        a[0] = arow[k];
        a[1] = arow[k + 1];
        b[0] = bcol[(size_t)k * N];
        b[1] = bcol[(size_t)(k + 1) * N];
        c = __builtin_amdgcn_wmma_f32_16x16x4_f32(false, a, false, b,
                                                  (short)0, c, false, false);
    }
    // D layout: lanes 0-15: vgpr r -> row r; lanes 16-31: vgpr r -> row 8+r.
    const int row0 = m0 + 8 * half;
    const int col  = n0 + lm;
#pragma unroll
    for (int r = 0; r < 8; ++r)
        C[(size_t)(row0 + r) * N + col] = c[r];
}

// ------------------------- edge scatter (propagation) ----------------------
// acc[dst] += t[src] * dinv[src]*dinv[dst], one thread per (edge, feature).
__global__ __launch_bounds__(256)
void k_propagate(const long long* __restrict__ src,
                 const long long* __restrict__ dst,
                 const float* __restrict__ dinv,
                 const float* __restrict__ t,
                 float* __restrict__ acc, int E) {
    long long tid = (long long)blockIdx.x * blockDim.x + threadIdx.x;
    int e = (int)(tid >> 6);
    int f = (int)(tid & 63);
    if (e >= E) return;
    int s = (int)src[e];
    int d = (int)dst[e];
    float coef = dinv[s] * dinv[d];
    atomicAdd(&acc[(size_t)d * FMID + f], t[(size_t)s * FMID + f] * coef);
}

// h = relu(acc1 + t1 * dinv^2 (self loop) + b1)
__global__ __launch_bounds__(256)
void k_finalize_h(const float* __restrict__ acc1, const float* __restrict__ t1,
                  const float* __restrict__ dinv, const float* __restrict__ b1,
                  float* __restrict__ h) {
    long long tid = (long long)blockIdx.x * blockDim.x + threadIdx.x;
    int i = (int)(tid >> 6);
    int f = (int)(tid & 63);
    if (i >= NNODES) return;
    float d2 = dinv[i] * dinv[i];
    size_t idx = (size_t)i * FMID + f;
    float v = acc1[idx] + t1[idx] * d2 + b1[f];
    h[idx] = fmaxf(v, 0.0f);
}

// mu = u[:, :32]+bmu, logstd = u[:, 32:]+bls,  z = mu + eps * exp(logstd)
__global__ __launch_bounds__(256)
void k_finalize_z(const float* __restrict__ acc2, const float* __restrict__ t2,
                  const float* __restrict__ dinv, const float* __restrict__ bmu,
                  const float* __restrict__ bls, const float* __restrict__ eps,
                  float* __restrict__ z) {
    long long tid = (long long)blockIdx.x * blockDim.x + threadIdx.x;
    int i = (int)(tid >> 5);
    int j = (int)(tid & 31);
    if (i >= NNODES) return;
    float d2 = dinv[i] * dinv[i];
    size_t b = (size_t)i * FMID;
    float mu = acc2[b + j]      + t2[b + j]      * d2 + bmu[j];
    float ls = acc2[b + j + 32] + t2[b + j + 32] * d2 + bls[j];
    z[(size_t)i * 32 + j] = mu + eps[(size_t)i * 32 + j] * __expf(ls);
}

// ------------------------- decoder: sigmoid(z z^T) -------------------------
// Symmetric: only tiles with tj >= ti are computed; off-diagonal tiles are
// additionally written transposed to the mirror location. Block = 256
// threads (8 waves) -> 128x128 output tile. z strips for the M and N sides
// are staged in LDS (stride 34 floats: conflict-free b64 frag reads). Wave w
// owns rows [w*16, w*16+16), sweeps 8 column subtiles with 8 f32 WMMAs each
// (K=32), then fused sigmoid + nontemporal stream-out. The mirror store
// transposes each 16x16 subtile through a per-wave LDS patch (stride 17 ->
// conflict-free; LDS ops within a wave are in-order, no barrier needed).
__global__ __launch_bounds__(256)
void k_decode(const float* __restrict__ z, float* __restrict__ adj) {
    const int ti = blockIdx.y;          // tile row
    const int tj = blockIdx.x;          // tile col
    if (tj < ti) return;                // symmetry: upper triangle only

    __shared__ float zM[128 * 34];
    __shared__ float zN[128 * 34];
    __shared__ float tp[8][16 * 17];    // per-wave transpose patch

    const int mBase = ti * 128;
    const int nBase = tj * 128;
    const int t = threadIdx.x;

    // Cooperative load: thread t stages 16 consecutive floats of one half-row.
    {
        const int row = t >> 1;
        const int col = (t & 1) * 16;
        const float4* gm = (const float4*)(z + (size_t)(mBase + row) * 32 + col);
        const float4* gn = (const float4*)(z + (size_t)(nBase + row) * 32 + col);
        float* sm = zM + row * 34 + col;
        float* sn = zN + row * 34 + col;
#pragma unroll
        for (int q = 0; q < 4; ++q) {
            float4 a = gm[q];
            float4 b = gn[q];
            ((float2*)(sm + 4 * q))[0] = make_float2(a.x, a.y);
            ((float2*)(sm + 4 * q))[1] = make_float2(a.z, a.w);
            ((float2*)(sn + 4 * q))[0] = make_float2(b.x, b.y);
            ((float2*)(sn + 4 * q))[1] = make_float2(b.z, b.w);
        }
    }
    __syncthreads();

    const int wave = t >> 5;
    const int lane = t & 31;
    const int half = lane >> 4;
    const int lm   = lane & 15;
    const int m0   = wave * 16;
    const bool mirror = (ti != tj);
    float* tpw = tp[wave];

    // Preload the 8 A-fragments for this wave's 16-row strip (reused 8x).
    v2f afrag[8];
    {
        const float* arow = zM + (m0 + lm) * 34 + 2 * half;
#pragma unroll
        for (int kk = 0; kk < 8; ++kk)
            afrag[kk] = *(const v2f*)(arow + 4 * kk);
    }

#pragma unroll
    for (int j = 0; j < 8; ++j) {
        const int n0 = j * 16;
        const float* brow = zN + (n0 + lm) * 34 + 2 * half;
        v8f c = {};
#pragma unroll
        for (int kk = 0; kk < 8; ++kk) {
            v2f b = *(const v2f*)(brow + 4 * kk);
            c = __builtin_amdgcn_wmma_f32_16x16x4_f32(false, afrag[kk], false,
                                                      b, (short)0, c,
                                                      false, false);
        }
        // Direct tile (ti, tj): sigmoid + nontemporal store.
        const int gcol  = nBase + n0 + lm;
        const int grow0 = mBase + m0 + 8 * half;
#pragma unroll
        for (int r = 0; r < 8; ++r) {
            float s = 1.0f / (1.0f + __expf(-c[r]));
            __builtin_nontemporal_store(s, adj + (size_t)(grow0 + r) * NNODES + gcol);
            if (mirror)   // stage S[row_local][col_local] for the transpose
                tpw[(r + 8 * half) * 17 + lm] = s;
        }
        // Mirror tile (tj, ti): read transposed from the patch, store
        // coalesced. In-wave LDS ordering makes this safe without a barrier.
        if (mirror) {
            const int mcol  = mBase + m0 + lm;
            const int mrow0 = nBase + n0 + 8 * half;
#pragma unroll
            for (int r = 0; r < 8; ++r) {
                float v = tpw[lm * 17 + r + 8 * half];
                __builtin_nontemporal_store(v, adj + (size_t)(mrow0 + r) * NNODES + mcol);
            }
        }
    }
}

// ------------------------------- launcher ----------------------------------

extern "C" void kernel_launch(void* const* d_in, const int* in_sizes, int n_in,
                              void* d_out, int out_size, void* d_ws, size_t ws_size,
                              hipStream_t stream) {
    const float*      x   = (const float*)d_in[0];
    const long long*  ei  = (const long long*)d_in[1];   // int64 per reference
    const float*      W1  = (const float*)d_in[2];
    const float*      b1  = (const float*)d_in[3];
    const float*      Wmu = (const float*)d_in[4];
    const float*      bmu = (const float*)d_in[5];
    const float*      Wls = (const float*)d_in[6];
    const float*      bls = (const float*)d_in[7];
    const float*      eps = (const float*)d_in[8];
    float*            adj = (float*)d_out;

    const int E = in_sizes[1] / 2;
    const long long* esrc = ei;
    const long long* edst = ei + E;

    // Workspace layout (all offsets 256B aligned; ~11.2 MB total).
    char* ws = (char*)d_ws;
    const size_t SZ_NF = (size_t)NNODES * FMID * sizeof(float);   // 2 MB
    float* acc1 = (float*)(ws);
    float* t1   = (float*)(ws + 1 * SZ_NF);
    float* h    = (float*)(ws + 2 * SZ_NF);
    float* t2   = (float*)(ws + 3 * SZ_NF);
    float* acc2 = (float*)(ws + 4 * SZ_NF);
    float* z    = (float*)(ws + 5 * SZ_NF);                       // 1 MB
    char*  tail = ws + 5 * SZ_NF + (size_t)NNODES * 32 * sizeof(float);
    float* deg  = (float*)(tail);
    float* dinv = (float*)(tail + NNODES * sizeof(float));
    float* Wcat = (float*)(tail + 2 * (size_t)NNODES * sizeof(float));

    const int nf = NNODES * FMID;

    // 0) init accumulators / degree (self-loops start deg at 1.0)
    k_fill<<<(nf + 255) / 256, 256, 0, stream>>>(acc1, 0.0f, nf);
    k_fill<<<(nf + 255) / 256, 256, 0, stream>>>(acc2, 0.0f, nf);
    k_fill<<<(NNODES + 255) / 256, 256, 0, stream>>>(deg, 1.0f, NNODES);

    // 1) degrees + D^{-1/2}
    k_degree<<<(E + 255) / 256, 256, 0, stream>>>(edst, deg, E);
    k_dinv<<<(NNODES + 255) / 256, 256, 0, stream>>>(deg, dinv, NNODES);

    // 2) pack [Wmu | Wls]
    k_pack_wcat<<<(64 * 64 + 255) / 256, 256, 0, stream>>>(Wmu, Wls, Wcat);

    // 3) layer 1: t1 = x @ W1 ; propagate ; h = relu(. + b1)
    {
        dim3 g(NNODES / 64, 64 / 16);
        k_gemm_wmma<<<g, 128, 0, stream>>>(x, W1, t1, NNODES, 64, 256);
    }
    {
        long long tot = (long long)E * FMID;
        k_propagate<<<(int)((tot + 255) / 256), 256, 0, stream>>>(esrc, edst, dinv, t1, acc1, E);
        k_finalize_h<<<(nf + 255) / 256, 256, 0, stream>>>(acc1, t1, dinv, b1, h);
    }

    // 4) layer 2 (both heads fused): t2 = h @ Wcat ; propagate ; z
    {
        dim3 g(NNODES / 64, 64 / 16);
        k_gemm_wmma<<<g, 128, 0, stream>>>(h, Wcat, t2, NNODES, 64, 64);
    }
    {
        long long tot = (long long)E * FMID;
        k_propagate<<<(int)((tot + 255) / 256), 256, 0, stream>>>(esrc, edst, dinv, t2, acc2, E);
        int nz = NNODES * 32;
        k_finalize_z<<<(nz + 255) / 256, 256, 0, stream>>>(acc2, t2, dinv, bmu, bls, eps, z);
    }

    // 5) decoder: adj = sigmoid(z z^T), symmetric 128x128 tiling
    {
        dim3 g(NNODES / 128, NNODES / 128);
        k_decode<<<g, 256, 0, stream>>>(z, adj);
    }
}